// MultiModalFusionModule_76982993814008
// MI455X (gfx1250) — compile-verified
//
#include <hip/hip_runtime.h>
#include <hip/hip_bf16.h>
#include <math.h>

// Shapes: B=16, C=256, H=W=128.  HW = 16384 elems = 4096 float4 per (b,c) slice.
// Elements per input = 16*256*16384 = 67,108,864 (= 16,777,216 float4).
#define HW4        4096      // float4 per channel slice
#define SLICES     4096      // b*c slices per input (16*256)
#define POOL_N     12288     // 3 * 4096 pooled sums in workspace
#define BATCH_F4   1048576   // float4 per batch (256*16384/4*... = 256*4096)
#define BLOCKS_PER_BATCH 256 // fuse kernel: 4096 f4 per block

typedef __attribute__((ext_vector_type(4))) float f4;
typedef __attribute__((ext_vector_type(2))) float v2f;
typedef __attribute__((ext_vector_type(8))) float v8f;

// ---------------------------------------------------------------------------
// Pass 1: pooled[j][b][c] = sum over H,W of x[b,c,:,:]
// One 256-thread block per (input j, batch b, channel c) slice (16384 floats,
// contiguous).  Deterministic tree reduction (shfl within wave32 + LDS).
// Non-temporal loads: 768 MB streamed once, never fits in 192 MB L2.
// ---------------------------------------------------------------------------
__global__ __launch_bounds__(256) void pool_kernel(
    const f4* __restrict__ gray, const f4* __restrict__ green,
    const f4* __restrict__ rgb, float* __restrict__ pooled)
{
    const int bid  = blockIdx.x;          // 0 .. 12287
    const int j    = bid >> 12;           // input index 0..2
    const int rem  = bid & 4095;          // (b,c) linear slice index
    const f4* src  = (j == 0) ? gray : ((j == 1) ? green : rgb);

    const size_t base = (size_t)rem * HW4;
    const int tid  = threadIdx.x;

    float s = 0.0f;
#pragma unroll
    for (int it = 0; it < 16; ++it) {
        f4 v = __builtin_nontemporal_load(&src[base + (size_t)it * 256 + tid]);
        s += (v.x + v.y) + (v.z + v.w);
    }
    // wave32 tree reduction
#pragma unroll
    for (int off = 16; off > 0; off >>= 1)
        s += __shfl_xor(s, off, 32);

    __shared__ float wsum[8];
    const int lane = tid & 31, wid = tid >> 5;
    if (lane == 0) wsum[wid] = s;
    __syncthreads();
    if (tid == 0) {
        float t = 0.0f;
#pragma unroll
        for (int i = 0; i < 8; ++i) t += wsum[i];
        pooled[bid] = t;    // pooled laid out [j][b*256 + c]
    }
}

// ---------------------------------------------------------------------------
// Pass 2: logits[b] = (pooled[b,:] . conv_w) / 16384 + conv_b ; scale = sigmoid.
// The einsum "bc,c->b" is a 16x256 mat-vec with M=16 == WMMA tile height:
// 64 chained V_WMMA_F32_16X16X4_F32 steps, B = conv_w broadcast along N.
// f32 A (16x4) layout (ISA 7.12.2): lanes 0-15 hold K=0,1 in V0,V1; lanes
// 16-31 hold K=2,3.  D column N=0 lives in lane 0 (M=0..7) / lane 16 (M=8..15).
// One wave per input tensor; folds the alpha=10 factor into the scale.
// ---------------------------------------------------------------------------
__global__ __launch_bounds__(32) void finalize_wmma(
    const float* __restrict__ pooled, const float* __restrict__ conv_w,
    const float* __restrict__ conv_b, float* __restrict__ scales)
{
    const int j    = blockIdx.x;          // input 0..2
    const int lane = threadIdx.x;         // wave32, EXEC all ones
    const int m    = lane & 15;           // batch row
    const int khl  = (lane >> 4) << 1;    // 0 for lanes<16, 2 for lanes>=16
    const float* P = pooled + j * 4096;   // [16][256]

    v8f acc = {};
#pragma unroll 4
    for (int t = 0; t < 64; ++t) {
        const int k0 = 4 * t + khl;
        v2f a, b;
        a.x = P[m * 256 + k0];
        a.y = P[m * 256 + k0 + 1];
        b.x = conv_w[k0];                 // broadcast along N
        b.y = conv_w[k0 + 1];
        acc = __builtin_amdgcn_wmma_f32_16x16x4_f32(
            /*neg_a=*/false, a, /*neg_b=*/false, b,
            /*c_mod=*/(short)0, acc, /*reuse_a=*/false, /*reuse_b=*/false);
    }

    const float factor = (j == 0) ? 1.0f : 10.0f;   // torch.add alpha=10
    const float bb = conv_b[0];
    if ((lane & 15) == 0) {               // lanes 0 and 16 hold column N=0
        const int mbase = (lane >> 4) * 8;
#pragma unroll
        for (int r = 0; r < 8; ++r) {
            const float logit = acc[r] * (1.0f / 16384.0f) + bb;
            scales[j * 16 + mbase + r] = factor / (1.0f + expf(-logit));
        }
    }
}

// ---------------------------------------------------------------------------
// Pass 3: out = gray*s_g + green*s_gn + rgb*s_rgb  (scales pre-multiplied).
// 1 GB streamed: 3 NT b128 loads + 1 NT b128 store per 4 elements.
// Batch index is block-uniform -> scale loads become s_load.
// ---------------------------------------------------------------------------
__global__ __launch_bounds__(256) void fuse_kernel(
    const f4* __restrict__ gray, const f4* __restrict__ green,
    const f4* __restrict__ rgb, const float* __restrict__ scales,
    f4* __restrict__ out)
{
    const int b  = blockIdx.x >> 8;       // BLOCKS_PER_BATCH = 256
    const float sg = scales[b];
    const float sn = scales[16 + b];
    const float sr = scales[32 + b];

    const size_t base = (size_t)blockIdx.x * 4096 + threadIdx.x;
#pragma unroll
    for (int it = 0; it < 16; ++it) {
        const size_t i = base + (size_t)it * 256;
        f4 a = __builtin_nontemporal_load(&gray[i]);
        f4 c = __builtin_nontemporal_load(&green[i]);
        f4 d = __builtin_nontemporal_load(&rgb[i]);
        f4 o;
        o.x = fmaf(d.x, sr, fmaf(c.x, sn, a.x * sg));
        o.y = fmaf(d.y, sr, fmaf(c.y, sn, a.y * sg));
        o.z = fmaf(d.z, sr, fmaf(c.z, sn, a.z * sg));
        o.w = fmaf(d.w, sr, fmaf(c.w, sn, a.w * sg));
        __builtin_nontemporal_store(o, &out[i]);
    }
}

// ---------------------------------------------------------------------------
extern "C" void kernel_launch(void* const* d_in, const int* in_sizes, int n_in,
                              void* d_out, int out_size, void* d_ws, size_t ws_size,
                              hipStream_t stream)
{
    const f4*    gray   = (const f4*)d_in[0];
    const f4*    green  = (const f4*)d_in[1];
    const f4*    rgb    = (const f4*)d_in[2];
    const float* conv_w = (const float*)d_in[3];
    const float* conv_b = (const float*)d_in[4];
    f4*          out    = (f4*)d_out;

    float* pooled = (float*)d_ws;          // 12288 floats
    float* scales = pooled + POOL_N;       // 48 floats: [j][b], alpha folded in

    pool_kernel<<<POOL_N, 256, 0, stream>>>(gray, green, rgb, pooled);
    finalize_wmma<<<3, 32, 0, stream>>>(pooled, conv_w, conv_b, scales);
    fuse_kernel<<<16 * BLOCKS_PER_BATCH, 256, 0, stream>>>(gray, green, rgb, scales, out);
}